// StructureExtractor_58437325030063
// MI455X (gfx1250) — compile-verified
//
#include <hip/hip_runtime.h>
#include <hip/hip_bf16.h>

// ---------------------------------------------------------------------------
// StructureExtractor on MI455X (gfx1250, wave32).
// Exact fp32 pipeline; GEMMs via V_WMMA_F32_16X16X4_F32 (fp32 matrix path).
// Edge segment-sums via hardware float atomics (global_atomic_add_f32).
// ---------------------------------------------------------------------------

typedef __attribute__((ext_vector_type(2))) float v2f;
typedef __attribute__((ext_vector_type(8))) float v8f;

#define D 128
#define LP1D 512          // (L+1)*D
#define KC 64             // K-chunk staged in LDS
#define LDSW 132          // padded LDS row stride (floats)

// ---------------------------------------------------------------------------
// Generic WMMA fp32 GEMM:  C[m,n] = f(A)[m, :K] @ W[:K, n] (+ bias[n])
//   f(A)[m,k] = A[m,k]*cscale[k] + cshift[k]   (identity if cscale==nullptr)
// One wave per 16-row strip, 8 column tiles of 16 (full 128-wide output).
// W staged through LDS in KC-row chunks.
// ---------------------------------------------------------------------------
__global__ __launch_bounds__(128)
void gemm_wmma_f32(const float* __restrict__ A, int lda,
                   const float* __restrict__ W,          // [K x 128] row-major
                   const float* __restrict__ cscale,     // [K] or nullptr
                   const float* __restrict__ cshift,     // [K] or nullptr
                   const float* __restrict__ bias,       // [128] or nullptr
                   float* __restrict__ C, int ldc,
                   int n_rows, int K)
{
    __shared__ float lw[KC * LDSW];

    const int tid    = threadIdx.x;
    const int wave   = tid >> 5;
    const int lane   = tid & 31;
    const int lane16 = lane & 15;
    const int hi     = lane >> 4;       // half-wave select
    const int koff   = hi * 2;          // K offset of this half-wave in 4-wide window

    const int strip  = blockIdx.x * 4 + wave;   // 16-row strip
    const int m_base = strip * 16;
    int m_ld = m_base + lane16;
    if (m_ld >= n_rows) m_ld = n_rows - 1;      // clamp loads; stores guarded

    v8f acc[8] = {};                            // 8 col tiles x (16x16 f32 C)

    for (int kb = 0; kb < K; kb += KC) {
        __syncthreads();
        // Stage W[kb:kb+KC, 0:128] into LDS (padded stride).
        for (int idx = tid; idx < KC * (D / 4); idx += 128) {
            const int r = idx >> 5;             // chunk row
            const int q = idx & 31;             // float4 within row
            const float4 w4 =
                ((const float4*)(W + (size_t)(kb + r) * D))[q];
            float* p = &lw[r * LDSW + q * 4];
            p[0] = w4.x; p[1] = w4.y; p[2] = w4.z; p[3] = w4.w;
        }
        __syncthreads();

        for (int ks = 0; ks < KC / 4; ++ks) {
            const int k0 = kb + ks * 4 + koff;
            // A operand (16x4 f32): lane m=lane16; VGPR0/1 = K=koff/koff+1.
            v2f a;
            a.x = A[(size_t)m_ld * lda + k0];
            a.y = A[(size_t)m_ld * lda + k0 + 1];
            if (cscale) {
                a.x = a.x * cscale[k0]     + cshift[k0];
                a.y = a.y * cscale[k0 + 1] + cshift[k0 + 1];
            }
            const int lrow = ks * 4 + koff;
#pragma unroll
            for (int ct = 0; ct < 8; ++ct) {
                const int n = ct * 16 + lane16;
                // B operand (4x16 f32): row K striped across lanes.
                v2f b;
                b.x = lw[lrow * LDSW + n];
                b.y = lw[(lrow + 1) * LDSW + n];
                acc[ct] = __builtin_amdgcn_wmma_f32_16x16x4_f32(
                    false, a, false, b, (short)0, acc[ct], false, false);
            }
        }
    }

    // C/D layout: lane -> n = lane16; VGPR j -> row m_base + j + hi*8.
#pragma unroll
    for (int ct = 0; ct < 8; ++ct) {
        const int n  = ct * 16 + lane16;
        const float bn = bias ? bias[n] : 0.0f;
#pragma unroll
        for (int j = 0; j < 8; ++j) {
            const int mr = m_base + j + hi * 8;
            if (mr < n_rows)
                C[(size_t)mr * ldc + n] = acc[ct][j] + bn;
        }
    }
}

// ---------------------------------------------------------------------------
// Utility / edge kernels
// ---------------------------------------------------------------------------
__global__ void zero_kernel(float* __restrict__ p, long long n)
{
    long long i = (long long)blockIdx.x * blockDim.x + threadIdx.x;
    if (i < n) p[i] = 0.0f;
}

__global__ void degree_kernel(const long long* __restrict__ src,
                              const long long* __restrict__ dst,
                              float* __restrict__ deg,
                              float* __restrict__ outc, long long E)
{
    long long e = (long long)blockIdx.x * blockDim.x + threadIdx.x;
    if (e >= E) return;
    atomicAdd(&deg[dst[e]],  1.0f);
    atomicAdd(&outc[src[e]], 1.0f);
}

__global__ void node_prep_kernel(const float* __restrict__ deg,
                                 float* __restrict__ outc,
                                 float* __restrict__ dinv,
                                 float* __restrict__ selfn, int N)
{
    int i = blockIdx.x * blockDim.x + threadIdx.x;
    if (i >= N) return;
    float r = rsqrtf(deg[i] + 1.0f);   // self-loop included
    dinv[i]  = r;
    selfn[i] = r * r;
    outc[i]  = fmaxf(outc[i], 1.0f);
}

__global__ void copy_x_kernel(const float* __restrict__ x,
                              float* __restrict__ xc, int N)
{
    long long idx = (long long)blockIdx.x * blockDim.x + threadIdx.x;
    if (idx >= (long long)N * D) return;
    long long i = idx >> 7;
    int dch     = (int)(idx & (D - 1));
    xc[i * LP1D + dch] = x[idx];
}

// agg[dst] += h[src] * (dinv[src]*dinv[dst]); one thread per (edge, 4 feats)
__global__ void edge_agg_kernel(const long long* __restrict__ src,
                                const long long* __restrict__ dst,
                                const float* __restrict__ h,
                                const float* __restrict__ dinv,
                                float* __restrict__ agg, long long E)
{
    long long t = (long long)blockIdx.x * blockDim.x + threadIdx.x;
    long long e = t >> 5;
    int g       = (int)(t & 31);
    if (e >= E) return;
    long long s = src[e], d = dst[e];
    float en = dinv[s] * dinv[d];
    const float4 hv = *(const float4*)(h + s * D + g * 4);
    float* ap = agg + d * D + g * 4;
    atomicAdd(ap + 0, hv.x * en);
    atomicAdd(ap + 1, hv.y * en);
    atomicAdd(ap + 2, hv.z * en);
    atomicAdd(ap + 3, hv.w * en);
}

// x_ = relu(agg + selfn*h + b)
__global__ void combine_kernel(const float* __restrict__ agg,
                               const float* __restrict__ h,
                               const float* __restrict__ selfn,
                               const float* __restrict__ b,
                               float* __restrict__ xl, int N)
{
    long long idx = (long long)blockIdx.x * blockDim.x + threadIdx.x;
    if (idx >= (long long)N * D) return;
    long long i = idx >> 7;
    int dch     = (int)(idx & (D - 1));
    float v = agg[idx] + selfn[i] * h[idx] + b[dch];
    xl[idx] = fmaxf(v, 0.0f);
}

// gsum[src] += (tau[src]-tau[dst])^2
__global__ void edge_gate_kernel(const long long* __restrict__ src,
                                 const long long* __restrict__ dst,
                                 const float* __restrict__ tau,
                                 float* __restrict__ gsum, long long E)
{
    long long t = (long long)blockIdx.x * blockDim.x + threadIdx.x;
    long long e = t >> 5;
    int g       = (int)(t & 31);
    if (e >= E) return;
    long long s = src[e], d = dst[e];
    const float4 ts = *(const float4*)(tau + s * D + g * 4);
    const float4 td = *(const float4*)(tau + d * D + g * 4);
    float4 q;
    q.x = (ts.x - td.x) * (ts.x - td.x);
    q.y = (ts.y - td.y) * (ts.y - td.y);
    q.z = (ts.z - td.z) * (ts.z - td.z);
    q.w = (ts.w - td.w) * (ts.w - td.w);
    float* gp = gsum + s * D + g * 4;
    atomicAdd(gp + 0, q.x);
    atomicAdd(gp + 1, q.y);
    atomicAdd(gp + 2, q.z);
    atomicAdd(gp + 3, q.w);
}

// gg = tanh(gsum/outc); x_new = (1-gg)*x_old + gg*x_ ; write into next xc slice
__global__ void update_kernel(const float* __restrict__ gsum,
                              const float* __restrict__ outc,
                              const float* __restrict__ xl,
                              float* __restrict__ xc, int layer, int N)
{
    long long idx = (long long)blockIdx.x * blockDim.x + threadIdx.x;
    if (idx >= (long long)N * D) return;
    long long i = idx >> 7;
    int dch     = (int)(idx & (D - 1));
    float gg = tanhf(gsum[idx] / outc[i]);
    float xo = xc[i * LP1D + (size_t)layer * D + dch];
    xc[i * LP1D + (size_t)(layer + 1) * D + dch] =
        (1.0f - gg) * xo + gg * xl[idx];
}

// BatchNorm stats: one thread per column, coalesced grid-stride over rows.
__global__ __launch_bounds__(LP1D)
void bn_sum_kernel(const float* __restrict__ xc,
                   float* __restrict__ sum, float* __restrict__ sumsq, int N)
{
    int c = threadIdx.x;                // 0..511
    float s = 0.0f, s2 = 0.0f;
    for (int r = blockIdx.x; r < N; r += gridDim.x) {
        float v = xc[(size_t)r * LP1D + c];
        s  += v;
        s2 += v * v;
    }
    atomicAdd(&sum[c],   s);
    atomicAdd(&sumsq[c], s2);
}

__global__ void bn_final_kernel(const float* __restrict__ sum,
                                const float* __restrict__ sumsq,
                                const float* __restrict__ gamma,
                                const float* __restrict__ beta,
                                float* __restrict__ scaleC,
                                float* __restrict__ shiftC, int N)
{
    int c = blockIdx.x * blockDim.x + threadIdx.x;
    if (c >= LP1D) return;
    float invN = 1.0f / (float)N;
    float mean = sum[c] * invN;
    float var  = sumsq[c] * invN - mean * mean;   // biased variance
    float sc   = gamma[c] * rsqrtf(var + 1e-5f);
    scaleC[c]  = sc;
    shiftC[c]  = beta[c] - mean * sc;
}

// ---------------------------------------------------------------------------
extern "C" void kernel_launch(void* const* d_in, const int* in_sizes, int n_in,
                              void* d_out, int out_size, void* d_ws, size_t ws_size,
                              hipStream_t stream)
{
    const float*     x     = (const float*)d_in[0];
    const long long* ei    = (const long long*)d_in[1];     // int64 [2,E]
    const float*     Ws    = (const float*)d_in[2];         // [3,128,128]
    const float*     bs    = (const float*)d_in[3];         // [3,128]
    const float*     gamma = (const float*)d_in[4];         // [512]
    const float*     beta  = (const float*)d_in[5];         // [512]
    const float*     Wout  = (const float*)d_in[6];         // [512,128]
    const float*     bout  = (const float*)d_in[7];         // [128]
    float*           out   = (float*)d_out;

    const int       N = in_sizes[0] / D;
    const long long E = (long long)in_sizes[1] / 2;
    const int       L = 3;

    const long long* src = ei;
    const long long* dst = ei + E;

    // Workspace layout (floats)
    float* w      = (float*)d_ws;
    size_t off    = 0;
    float* xc     = w + off; off += (size_t)N * LP1D;   // concat buffer
    float* h      = w + off; off += (size_t)N * D;
    float* agg    = w + off; off += (size_t)N * D;
    float* xl     = w + off; off += (size_t)N * D;      // x_ == tau
    float* gsum   = w + off; off += (size_t)N * D;
    float* deg    = w + off; off += (size_t)N;
    float* outc   = w + off; off += (size_t)N;
    float* dinv   = w + off; off += (size_t)N;
    float* selfn  = w + off; off += (size_t)N;
    float* sum    = w + off; off += LP1D;
    float* sumsq  = w + off; off += LP1D;
    float* scaleC = w + off; off += LP1D;
    float* shiftC = w + off; off += LP1D;

    const int TB = 256;
    auto blocks  = [](long long n, int tb) { return (unsigned)((n + tb - 1) / tb); };

    const long long ND = (long long)N * D;
    const long long ET = E * 32;                    // edge threads (4 feats each)
    const unsigned gemm_blocks = (unsigned)(((N + 15) / 16 + 3) / 4);

    // Degrees and normalization terms
    zero_kernel<<<blocks(N, TB), TB, 0, stream>>>(deg, N);
    zero_kernel<<<blocks(N, TB), TB, 0, stream>>>(outc, N);
    degree_kernel<<<blocks(E, TB), TB, 0, stream>>>(src, dst, deg, outc, E);
    node_prep_kernel<<<blocks(N, TB), TB, 0, stream>>>(deg, outc, dinv, selfn, N);
    copy_x_kernel<<<blocks(ND, TB), TB, 0, stream>>>(x, xc, N);

    for (int l = 0; l < L; ++l) {
        const float* xcur = xc + (size_t)l * D;     // row stride LP1D
        // h = xcur @ Ws[l]
        gemm_wmma_f32<<<gemm_blocks, 128, 0, stream>>>(
            xcur, LP1D, Ws + (size_t)l * D * D,
            nullptr, nullptr, nullptr, h, D, N, D);
        // agg = segment_sum(h[src]*enorm, dst)
        zero_kernel<<<blocks(ND, TB), TB, 0, stream>>>(agg, ND);
        edge_agg_kernel<<<blocks(ET, TB), TB, 0, stream>>>(src, dst, h, dinv, agg, E);
        // x_ = relu(agg + selfn*h + bs[l])
        combine_kernel<<<blocks(ND, TB), TB, 0, stream>>>(agg, h, selfn,
                                                          bs + (size_t)l * D, xl, N);
        // gate: gsum = segment_sum((tau[src]-tau[dst])^2, src)
        zero_kernel<<<blocks(ND, TB), TB, 0, stream>>>(gsum, ND);
        edge_gate_kernel<<<blocks(ET, TB), TB, 0, stream>>>(src, dst, xl, gsum, E);
        // x = (1-gg)*x + gg*x_  -> xc slice l+1
        update_kernel<<<blocks(ND, TB), TB, 0, stream>>>(gsum, outc, xl, xc, l, N);
    }

    // BatchNorm (training stats) folded into per-column affine
    zero_kernel<<<blocks(LP1D, TB), TB, 0, stream>>>(sum, LP1D);
    zero_kernel<<<blocks(LP1D, TB), TB, 0, stream>>>(sumsq, LP1D);
    bn_sum_kernel<<<256, LP1D, 0, stream>>>(xc, sum, sumsq, N);
    bn_final_kernel<<<blocks(LP1D, TB), TB, 0, stream>>>(sum, sumsq, gamma, beta,
                                                         scaleC, shiftC, N);

    // out = ((xc - mean)*inv*gamma + beta) @ Wout + bout
    gemm_wmma_f32<<<gemm_blocks, 128, 0, stream>>>(
        xc, LP1D, Wout, scaleC, shiftC, bout, out, D, N, LP1D);
}